// GATv2_16673063043494
// MI455X (gfx1250) — compile-verified
//
#include <hip/hip_runtime.h>
#include <math.h>

#define N_NODES 50000
#define N_EDGES 600000
#define N_GRAPHS 128
#define DIM_IN  128
#define HC1     128   // heads*dim_h for layer 1
#define H1      2
#define C1      64
#define DIM_OUT 64
#define E_TOT   (N_EDGES + N_NODES)   // with self loops
#define NEG_SLOPE 0.2f

typedef __attribute__((ext_vector_type(2))) float v2f;
typedef __attribute__((ext_vector_type(8))) float v8f;

// ---------------------------------------------------------------- utilities
__global__ void fill_f32(float* p, float v, int n) {
    int i = blockIdx.x * blockDim.x + threadIdx.x;
    if (i < n) p[i] = v;
}

__device__ __forceinline__ float atomicMaxF(float* addr, float val) {
    // classic monotone bit-pattern trick; buffer must be initialized to -inf
    if (val >= 0.0f)
        return __int_as_float(atomicMax((int*)addr, __float_as_int(val)));
    else
        return __uint_as_float(atomicMin((unsigned int*)addr, __float_as_uint(val)));
}

// ------------------------------------------------- WMMA fp32 GEMM (+ bias)
// C[M,N] = A[M,K] @ W[K,N] + bias[N],  N == NT*16, K % 4 == 0.
// Register-blocked: each wave owns 16 rows x full N (NT accumulators), so one
// A-fragment load per K-step feeds NT WMMAs. 5 waves/block x 625 blocks = 3125
// row tiles = M/16 exactly -> no guards, EXEC all-ones for every WMMA.
// A frag (16x4): lanes 0-15 row M=lane hold K=k,k+1; lanes 16-31 hold K=k+2,k+3.
// B frag (4x16): lane%16 = column; lanes<16 hold K=k,k+1; lanes>=16 K=k+2,k+3.
template <int NT>
__global__ void gemm_bias_wmma(const float* __restrict__ A,
                               const float* __restrict__ W,
                               const float* __restrict__ bias,
                               float* __restrict__ C,
                               int M, int K) {
    const int N = NT * 16;
    const int lane = threadIdx.x & 31;
    const int wave = threadIdx.x >> 5;
    const int m0 = (blockIdx.x * (blockDim.x >> 5) + wave) * 16;
    const int lr = lane & 15;
    const int ko = (lane & 16) ? 2 : 0;

    const float* ap = A + (size_t)(m0 + lr) * K + ko;   // two contiguous K
    const float* bp = W + (size_t)ko * N + lr;          // two rows of W

    v8f acc[NT];
#pragma unroll
    for (int t = 0; t < NT; ++t) acc[t] = (v8f){};

    for (int k = 0; k < K; k += 4) {
        v2f a;
        a.x = ap[k];
        a.y = ap[k + 1];
#pragma unroll
        for (int t = 0; t < NT; ++t) {
            v2f b;
            b.x = bp[(size_t)k * N + t * 16];
            b.y = bp[(size_t)(k + 1) * N + t * 16];
            acc[t] = __builtin_amdgcn_wmma_f32_16x16x4_f32(
                false, a, false, b, (short)0, acc[t], false, false);
        }
    }

    const int rbase = m0 + ((lane & 16) ? 8 : 0);
#pragma unroll
    for (int t = 0; t < NT; ++t) {
        const float bv = bias[t * 16 + lr];
#pragma unroll
        for (int v = 0; v < 8; ++v)
            C[(size_t)(rbase + v) * N + t * 16 + lr] = acc[t][v] + bv;
    }
}

// ---------------------------------------------- edge pass 1: logits + max
// One wave per edge; each lane covers 4 contiguous channels (float4 loads).
// HC=128 -> 32 lanes x 4; HC=64 -> lanes 16-31 contribute zeros.
__global__ void edge_logits(const float* __restrict__ xl,
                            const float* __restrict__ xr,
                            const int* __restrict__ src,
                            const int* __restrict__ dst,
                            const float* __restrict__ att,
                            float* __restrict__ logits,
                            float* __restrict__ maxb,
                            int H, int C, int HC) {
    const int wave = threadIdx.x >> 5;
    const int lane = threadIdx.x & 31;
    const int e = blockIdx.x * (blockDim.x >> 5) + wave;
    if (e >= E_TOT) return;
    const int s = (e < N_EDGES) ? src[e] : (e - N_EDGES);
    const int d = (e < N_EDGES) ? dst[e] : (e - N_EDGES);

    float part[2] = {0.0f, 0.0f};   // up to 2 heads
    for (int c0 = lane * 4; c0 < HC; c0 += 128) {
        const float4 l  = *(const float4*)(xl + (size_t)s * HC + c0);
        const float4 r  = *(const float4*)(xr + (size_t)d * HC + c0);
        const float4 at = *(const float4*)(att + c0);
        float v0 = l.x + r.x; v0 = (v0 > 0.0f) ? v0 : v0 * NEG_SLOPE;
        float v1 = l.y + r.y; v1 = (v1 > 0.0f) ? v1 : v1 * NEG_SLOPE;
        float v2 = l.z + r.z; v2 = (v2 > 0.0f) ? v2 : v2 * NEG_SLOPE;
        float v3 = l.w + r.w; v3 = (v3 > 0.0f) ? v3 : v3 * NEG_SLOPE;
        part[c0 / C] += v0 * at.x + v1 * at.y + v2 * at.z + v3 * at.w;
    }
    for (int h = 0; h < H; ++h) {
        float x = part[h];
        for (int off = 16; off >= 1; off >>= 1)
            x += __shfl_xor(x, off, 32);
        if (lane == 0) {
            logits[(size_t)e * H + h] = x;
            atomicMaxF(&maxb[(size_t)d * H + h], x);
        }
    }
}

// ------------------------------- edge pass 2: exp + scatter-accumulate
// One block per edge, blockDim.x == HC/4; each thread: 1 float4 load + 4
// f32 L2 atomics. Accumulates UNNORMALIZED sums; per-node divide later.
__global__ void edge_accum(const float* __restrict__ xl,
                           const int* __restrict__ src,
                           const int* __restrict__ dst,
                           const float* __restrict__ logits,
                           const float* __restrict__ maxb,
                           float* __restrict__ acc,
                           float* __restrict__ denom,
                           int H, int C, int HC) {
    const int e = blockIdx.x;
    const int c0 = threadIdx.x * 4;
    const int s = (e < N_EDGES) ? src[e] : (e - N_EDGES);
    const int d = (e < N_EDGES) ? dst[e] : (e - N_EDGES);
    const int h = c0 / C;
    const float a = __expf(logits[(size_t)e * H + h] - maxb[(size_t)d * H + h]);
    const float4 v = *(const float4*)(xl + (size_t)s * HC + c0);
    float* dstp = acc + (size_t)d * HC + c0;
    atomicAdd(dstp + 0, a * v.x);
    atomicAdd(dstp + 1, a * v.y);
    atomicAdd(dstp + 2, a * v.z);
    atomicAdd(dstp + 3, a * v.w);
    if ((c0 % C) == 0) atomicAdd(&denom[(size_t)d * H + h], a);
}

// -------------------------- per-node: normalize + bias + ELU (in place)
__global__ void finalize_elu(float* __restrict__ acc,
                             const float* __restrict__ denom,
                             const float* __restrict__ bias,
                             int N, int H, int C, int HC) {
    const int i = blockIdx.x * blockDim.x + threadIdx.x;
    if (i >= N * HC) return;
    const int n = i / HC;
    const int c = i % HC;
    const int h = c / C;
    float v = acc[i] / (denom[(size_t)n * H + h] + 1e-16f) + bias[c];
    acc[i] = (v > 0.0f) ? v : (__expf(v) - 1.0f);
}

// ------------------------------------------- BatchNorm (training stats)
__global__ void bn_stats(const float* __restrict__ h, float* __restrict__ stats,
                         int N, int HC) {
    __shared__ float ss[256], sq[256];
    const int c = blockIdx.x;
    float s = 0.0f, s2 = 0.0f;
    for (int n = threadIdx.x; n < N; n += blockDim.x) {
        float v = h[(size_t)n * HC + c];
        s += v; s2 += v * v;
    }
    ss[threadIdx.x] = s; sq[threadIdx.x] = s2;
    __syncthreads();
    for (int off = blockDim.x >> 1; off > 0; off >>= 1) {
        if ((int)threadIdx.x < off) {
            ss[threadIdx.x] += ss[threadIdx.x + off];
            sq[threadIdx.x] += sq[threadIdx.x + off];
        }
        __syncthreads();
    }
    if (threadIdx.x == 0) {
        float mu = ss[0] / (float)N;
        stats[c] = mu;
        stats[HC + c] = sq[0] / (float)N - mu * mu;  // biased var, matches jnp.var
    }
}

__global__ void bn_apply(float* __restrict__ h, const float* __restrict__ stats,
                         const float* __restrict__ gamma,
                         const float* __restrict__ beta, int N, int HC) {
    const int i = blockIdx.x * blockDim.x + threadIdx.x;
    if (i >= N * HC) return;
    const int c = i % HC;
    h[i] = (h[i] - stats[c]) * rsqrtf(stats[HC + c] + 1e-5f) * gamma[c] + beta[c];
}

// -------------------------------------------------- global mean pool
__global__ void pool_kernel(const float* __restrict__ h2,
                            const int* __restrict__ batch,
                            float* __restrict__ pooled, float* __restrict__ cnt,
                            int N, int D) {
    const int i = blockIdx.x * blockDim.x + threadIdx.x;
    if (i >= N * D) return;
    const int n = i / D;
    const int c = i % D;
    const int g = batch[n];
    atomicAdd(&pooled[(size_t)g * D + c], h2[i]);
    if (c == 0) atomicAdd(&cnt[g], 1.0f);
}

__global__ void final_linear(const float* __restrict__ pooled,
                             const float* __restrict__ cnt,
                             const float* __restrict__ Wlin,
                             const float* __restrict__ blin,
                             float* __restrict__ out) {
    const int g = blockIdx.x;
    const int j = threadIdx.x;
    if (j >= 2) return;
    const float inv = 1.0f / fmaxf(cnt[g], 1.0f);
    float s = blin[j];
    for (int k = 0; k < DIM_OUT; ++k)
        s += pooled[(size_t)g * DIM_OUT + k] * inv * Wlin[k * 2 + j];
    out[g * 2 + j] = s;
}

// ================================================================ launch
extern "C" void kernel_launch(void* const* d_in, const int* in_sizes, int n_in,
                              void* d_out, int out_size, void* d_ws, size_t ws_size,
                              hipStream_t stream) {
    const float* x     = (const float*)d_in[0];
    const float* Wl1   = (const float*)d_in[1];
    const float* bl1   = (const float*)d_in[2];
    const float* Wr1   = (const float*)d_in[3];
    const float* br1   = (const float*)d_in[4];
    const float* att1  = (const float*)d_in[5];
    const float* b1    = (const float*)d_in[6];
    const float* gamma = (const float*)d_in[7];
    const float* beta  = (const float*)d_in[8];
    const float* Wl2   = (const float*)d_in[9];
    const float* bl2   = (const float*)d_in[10];
    const float* Wr2   = (const float*)d_in[11];
    const float* br2   = (const float*)d_in[12];
    const float* att2  = (const float*)d_in[13];
    const float* b2    = (const float*)d_in[14];
    const float* Wlin  = (const float*)d_in[15];
    const float* blin  = (const float*)d_in[16];
    const int*   eidx  = (const int*)d_in[17];
    const int*   batch = (const int*)d_in[18];
    const int* src = eidx;            // edge_index[0]
    const int* dst = eidx + N_EDGES;  // edge_index[1]
    float* out = (float*)d_out;

    // ---- workspace layout (floats); regions reused across layers
    float* ws = (float*)d_ws;
    size_t o = 0;
    float* xl     = ws + o; o += (size_t)N_NODES * HC1;   // layer2 xl reuses
    float* xr     = ws + o; o += (size_t)N_NODES * HC1;   // layer2 xr reuses
    float* accb   = ws + o; o += (size_t)N_NODES * HC1;   // h1 -> hbn -> acc2 -> h2
    float* logits = ws + o; o += (size_t)E_TOT * H1;
    float* maxb   = ws + o; o += (size_t)N_NODES * H1;
    float* denb   = ws + o; o += (size_t)N_NODES * H1;
    float* stats  = ws + o; o += 2 * HC1;
    float* pooled = ws + o; o += (size_t)N_GRAPHS * DIM_OUT;
    float* cnt    = ws + o; o += N_GRAPHS;
    (void)ws_size; (void)in_sizes; (void)n_in; (void)out_size;

    const int GB = 625;                        // gemm blocks: 625 * 5 waves = 3125 tiles
    const int GT = 160;                        // 5 waves per block
    const int EB = (E_TOT + 7) / 8;            // edge_logits blocks (8 waves/blk)

    // =================== layer 1 ===================
    gemm_bias_wmma<HC1 / 16><<<dim3(GB), dim3(GT), 0, stream>>>(x, Wl1, bl1, xl, N_NODES, DIM_IN);
    gemm_bias_wmma<HC1 / 16><<<dim3(GB), dim3(GT), 0, stream>>>(x, Wr1, br1, xr, N_NODES, DIM_IN);

    fill_f32<<<dim3((N_NODES * H1 + 255) / 256), dim3(256), 0, stream>>>(maxb, -INFINITY, N_NODES * H1);
    fill_f32<<<dim3((N_NODES * H1 + 255) / 256), dim3(256), 0, stream>>>(denb, 0.0f, N_NODES * H1);
    fill_f32<<<dim3((N_NODES * HC1 + 255) / 256), dim3(256), 0, stream>>>(accb, 0.0f, N_NODES * HC1);

    edge_logits<<<dim3(EB), dim3(256), 0, stream>>>(xl, xr, src, dst, att1, logits, maxb, H1, C1, HC1);
    edge_accum<<<dim3(E_TOT), dim3(HC1 / 4), 0, stream>>>(xl, src, dst, logits, maxb, accb, denb, H1, C1, HC1);
    finalize_elu<<<dim3((N_NODES * HC1 + 255) / 256), dim3(256), 0, stream>>>(accb, denb, b1, N_NODES, H1, C1, HC1);

    bn_stats<<<dim3(HC1), dim3(256), 0, stream>>>(accb, stats, N_NODES, HC1);
    bn_apply<<<dim3((N_NODES * HC1 + 255) / 256), dim3(256), 0, stream>>>(accb, stats, gamma, beta, N_NODES, HC1);

    // =================== layer 2 ===================
    gemm_bias_wmma<DIM_OUT / 16><<<dim3(GB), dim3(GT), 0, stream>>>(accb, Wl2, bl2, xl, N_NODES, HC1);
    gemm_bias_wmma<DIM_OUT / 16><<<dim3(GB), dim3(GT), 0, stream>>>(accb, Wr2, br2, xr, N_NODES, HC1);

    fill_f32<<<dim3((N_NODES + 255) / 256), dim3(256), 0, stream>>>(maxb, -INFINITY, N_NODES);
    fill_f32<<<dim3((N_NODES + 255) / 256), dim3(256), 0, stream>>>(denb, 0.0f, N_NODES);
    fill_f32<<<dim3((N_NODES * DIM_OUT + 255) / 256), dim3(256), 0, stream>>>(accb, 0.0f, N_NODES * DIM_OUT);

    edge_logits<<<dim3(EB), dim3(256), 0, stream>>>(xl, xr, src, dst, att2, logits, maxb, 1, DIM_OUT, DIM_OUT);
    edge_accum<<<dim3(E_TOT), dim3(DIM_OUT / 4), 0, stream>>>(xl, src, dst, logits, maxb, accb, denb, 1, DIM_OUT, DIM_OUT);
    finalize_elu<<<dim3((N_NODES * DIM_OUT + 255) / 256), dim3(256), 0, stream>>>(accb, denb, b2, N_NODES, 1, DIM_OUT, DIM_OUT);

    // =================== pool + head ===================
    fill_f32<<<dim3((N_GRAPHS * DIM_OUT + 255) / 256), dim3(256), 0, stream>>>(pooled, 0.0f, N_GRAPHS * DIM_OUT);
    fill_f32<<<dim3(1), dim3(256), 0, stream>>>(cnt, 0.0f, N_GRAPHS);
    pool_kernel<<<dim3((N_NODES * DIM_OUT + 255) / 256), dim3(256), 0, stream>>>(accb, batch, pooled, cnt, N_NODES, DIM_OUT);
    final_linear<<<dim3(N_GRAPHS), dim3(32), 0, stream>>>(pooled, cnt, Wlin, blin, out);
}